// GraphNetwork_44555990729424
// MI455X (gfx1250) — compile-verified
//
#include <hip/hip_runtime.h>
#include <math.h>

// Problem constants (match reference)
#define T_   16
#define N_   128
#define F_   128
#define ROWS (T_ * N_)   // 2048
#define FQ_  (F_ / 4)    // 32 float4 per feature row

typedef __attribute__((ext_vector_type(2))) float v2f;
typedef __attribute__((ext_vector_type(8))) float v8f;

__device__ __forceinline__ float lrelu(float v) { return v >= 0.0f ? v : 0.01f * v; }

#if __has_builtin(__builtin_amdgcn_tanhf)
__device__ __forceinline__ float fast_tanh(float x) { return __builtin_amdgcn_tanhf(x); }
#else
__device__ __forceinline__ float fast_tanh(float x) { return tanhf(x); }
#endif

// ---------------------------------------------------------------------------
// Row L2-norms: one wave per row (32 lanes x float4 = 128 floats), shfl reduce
// grid = ROWS/8, block = 256 (8 waves)
// ---------------------------------------------------------------------------
__global__ void row_norm_kernel(const float* __restrict__ x, float* __restrict__ nrm) {
    int wave = threadIdx.x >> 5;
    int lane = threadIdx.x & 31;
    int row  = blockIdx.x * 8 + wave;
    const float4* p = reinterpret_cast<const float4*>(x + (size_t)row * F_);
    float4 v = p[lane];
    float s = v.x * v.x + v.y * v.y + v.z * v.z + v.w * v.w;
    for (int off = 16; off > 0; off >>= 1) s += __shfl_xor(s, off, 32);
    if (lane == 0) nrm[row] = sqrtf(s);
}

// ---------------------------------------------------------------------------
// Cosine-similarity via WMMA f32 16x16x4: per-task gram G = x x^T, fused
// epilogue computes clip(sim), clip(1-sim). One wave per 16x16 tile.
// grid = 128, block = 256 (8 waves) -> 1024 tiles = 16 tasks * 8*8 tiles
// ---------------------------------------------------------------------------
__global__ void sim_wmma_kernel(const float* __restrict__ x, const float* __restrict__ nrm,
                                float* __restrict__ out) {
    int wave = threadIdx.x >> 5;
    int lane = threadIdx.x & 31;
    int tile = blockIdx.x * 8 + wave;        // 0..1023
    int t    = tile >> 6;                    // task
    int rem  = tile & 63;
    int m0   = (rem >> 3) * 16;
    int n0   = (rem & 7) * 16;
    int lp   = lane & 15;
    int half = lane >> 4;

    const float* xt   = x + (size_t)t * N_ * F_;
    const float* arow = xt + (size_t)(m0 + lp) * F_;
    const float* brow = xt + (size_t)(n0 + lp) * F_;

    v8f acc = {};
    for (int k0 = 0; k0 < F_; k0 += 4) {
        int kk = k0 + 2 * half;              // VGPR0:K={0,2}, VGPR1:K={1,3}
        v2f a, b;
        a.x = arow[kk]; a.y = arow[kk + 1];
        b.x = brow[kk]; b.y = brow[kk + 1];  // B[k][n] = x[n][k]: transpose folded into layout
        acc = __builtin_amdgcn_wmma_f32_16x16x4_f32(false, a, false, b, (short)0, acc,
                                                    false, false);
    }

    const float* nt = nrm + t * N_;
    float nj = nt[n0 + lp];
    float* o0 = out + (size_t)t * 2 * N_ * N_;
    float* o1 = o0 + (size_t)N_ * N_;
    for (int v = 0; v < 8; ++v) {
        int m = m0 + v + 8 * half;           // D layout: VGPR v -> M = v + 8*half, N = lane%16
        float s  = acc[v] / (nt[m] * nj + 1e-6f);
        o0[(size_t)m * N_ + n0 + lp] = fminf(fmaxf(s, 0.0f), 1.0f);
        o1[(size_t)m * N_ + n0 + lp] = fminf(fmaxf(1.0f - s, 0.0f), 1.0f);
    }
}

// ---------------------------------------------------------------------------
// Fused edge update + aggregation, float4-vectorized (global_load_b128).
// Block = 256 threads: 32 feature-quad lanes x 8 parallel j-slices; each
// thread runs 16 iterations of j. layer==1 recomputes e0 from the original
// edge input so layer-0 edges are never materialized (saves 134MB wr + rd).
// grid = ROWS (one block per (t,i)).
// ---------------------------------------------------------------------------
__global__ void edge_aggr_kernel(const float* __restrict__ x0, const float* __restrict__ xl,
                                 const float* __restrict__ E,
                                 const float* __restrict__ edge_w, const float* __restrict__ edge_b,
                                 float* __restrict__ aggr, int layer) {
    __shared__ float4 part[256];
    int tid = threadIdx.x;
    int fq  = tid & 31;                      // features 4*fq .. 4*fq+3
    int jp  = tid >> 5;                      // 0..7
    int row = blockIdx.x;                    // t*N + i
    int t   = row >> 7;
    int i   = row & (N_ - 1);

    float w00 = edge_w[0], w01 = edge_w[1], w02 = edge_w[2], b0 = edge_b[0];
    float w10 = edge_w[3], w11 = edge_w[4], w12 = edge_w[5], b1 = edge_b[1];

    const float4* x0t  = reinterpret_cast<const float4*>(x0 + (size_t)t * N_ * F_);
    const float4* xlt  = reinterpret_cast<const float4*>(xl + (size_t)t * N_ * F_);
    const float4* Erow = reinterpret_cast<const float4*>(E + (size_t)row * N_ * F_);

    float4 x0i = x0t[(size_t)i * FQ_ + fq];
    float4 xli = xlt[(size_t)i * FQ_ + fq];

    float4 acc = make_float4(0.0f, 0.0f, 0.0f, 0.0f);
    for (int j = jp; j < N_; j += 8) {
        __builtin_prefetch(Erow + (size_t)(j + 8) * FQ_ + fq, 0, 0);
        if (j == i) continue;                // diagonal is masked to zero
        float4 Ev = Erow[(size_t)j * FQ_ + fq];
        if (layer == 0) {
            float4 xj = xlt[(size_t)j * FQ_ + fq];
            acc.x += fast_tanh(w00 * xli.x + w01 * (xli.x - xj.x) + w02 * Ev.x + b0);
            acc.y += fast_tanh(w00 * xli.y + w01 * (xli.y - xj.y) + w02 * Ev.y + b0);
            acc.z += fast_tanh(w00 * xli.z + w01 * (xli.z - xj.z) + w02 * Ev.z + b0);
            acc.w += fast_tanh(w00 * xli.w + w01 * (xli.w - xj.w) + w02 * Ev.w + b0);
        } else {
            float4 x0j = x0t[(size_t)j * FQ_ + fq];
            float4 x1j = xlt[(size_t)j * FQ_ + fq];
            float e0;
            e0 = fast_tanh(w00 * x0i.x + w01 * (x0i.x - x0j.x) + w02 * Ev.x + b0);
            acc.x += fast_tanh(w10 * xli.x + w11 * (xli.x - x1j.x) + w12 * e0 + b1);
            e0 = fast_tanh(w00 * x0i.y + w01 * (x0i.y - x0j.y) + w02 * Ev.y + b0);
            acc.y += fast_tanh(w10 * xli.y + w11 * (xli.y - x1j.y) + w12 * e0 + b1);
            e0 = fast_tanh(w00 * x0i.z + w01 * (x0i.z - x0j.z) + w02 * Ev.z + b0);
            acc.z += fast_tanh(w10 * xli.z + w11 * (xli.z - x1j.z) + w12 * e0 + b1);
            e0 = fast_tanh(w00 * x0i.w + w01 * (x0i.w - x0j.w) + w02 * Ev.w + b0);
            acc.w += fast_tanh(w10 * xli.w + w11 * (xli.w - x1j.w) + w12 * e0 + b1);
        }
    }
    part[tid] = acc;
    __syncthreads();
    if (tid < 32) {                          // 8-way reduce over j-slices
        float4 s = part[tid];
        for (int k = 1; k < 8; ++k) {
            float4 p = part[k * 32 + tid];
            s.x += p.x; s.y += p.y; s.z += p.z; s.w += p.w;
        }
        reinterpret_cast<float4*>(aggr + (size_t)row * F_)[tid] = s;
    }
}

// ---------------------------------------------------------------------------
// y = lrelu(c0*x + c1*aggr) @ W^T via WMMA f32 16x16x4 (h fused into A-frag).
// One wave per 16x16 tile of y [2048 x 128]. grid = 128, block = 256.
// ---------------------------------------------------------------------------
__global__ void node_gemm_kernel(const float* __restrict__ xl, const float* __restrict__ aggr,
                                 const float* __restrict__ comb_w, const float* __restrict__ node_w,
                                 float* __restrict__ y, int layer) {
    int wave = threadIdx.x >> 5;
    int lane = threadIdx.x & 31;
    int tile = blockIdx.x * 8 + wave;        // 0..1023
    int m0   = (tile >> 3) * 16;             // 128 row-tiles
    int n0   = (tile & 7) * 16;              // 8 col-tiles
    int lp   = lane & 15;
    int half = lane >> 4;

    float c0 = comb_w[layer * 2 + 0];
    float c1 = comb_w[layer * 2 + 1];
    const float* W  = node_w + (size_t)layer * F_ * F_;
    const float* ax = xl   + (size_t)(m0 + lp) * F_;
    const float* ag = aggr + (size_t)(m0 + lp) * F_;
    const float* bw = W    + (size_t)(n0 + lp) * F_;   // B[k][g] = W[g][k]

    v8f acc = {};
    for (int k0 = 0; k0 < F_; k0 += 4) {
        int kk = k0 + 2 * half;
        v2f a, b;
        a.x = lrelu(c0 * ax[kk]     + c1 * ag[kk]);
        a.y = lrelu(c0 * ax[kk + 1] + c1 * ag[kk + 1]);
        b.x = bw[kk]; b.y = bw[kk + 1];
        acc = __builtin_amdgcn_wmma_f32_16x16x4_f32(false, a, false, b, (short)0, acc,
                                                    false, false);
    }
    for (int v = 0; v < 8; ++v)
        y[(size_t)(m0 + v + 8 * half) * F_ + n0 + lp] = acc[v];
}

// ---------------------------------------------------------------------------
// BatchNorm stats over (T,N) = 2048 rows per feature; deterministic block
// reduction; emits fused scale/shift. grid = F_ (one block per feature).
// ---------------------------------------------------------------------------
__global__ void bn_stats_kernel(const float* __restrict__ y,
                                const float* __restrict__ gamma, const float* __restrict__ beta,
                                float* __restrict__ scale, float* __restrict__ shift, int layer) {
    __shared__ float s1[256];
    __shared__ float s2[256];
    int g   = blockIdx.x;
    int tid = threadIdx.x;
    float a = 0.0f, b = 0.0f;
    for (int r = tid; r < ROWS; r += 256) {
        float v = y[(size_t)r * F_ + g];
        a += v; b += v * v;
    }
    s1[tid] = a; s2[tid] = b;
    __syncthreads();
    for (int off = 128; off > 0; off >>= 1) {
        if (tid < off) { s1[tid] += s1[tid + off]; s2[tid] += s2[tid + off]; }
        __syncthreads();
    }
    if (tid == 0) {
        float mean = s1[0] * (1.0f / ROWS);
        float var  = s2[0] * (1.0f / ROWS) - mean * mean;   // population var (ddof=0)
        float sc   = gamma[layer * F_ + g] * rsqrtf(var + 1e-5f);
        scale[g] = sc;
        shift[g] = beta[layer * F_ + g] - mean * sc;
    }
}

// Normalize + LeakyReLU -> next layer node feats, float4-vectorized.
// grid = (ROWS*F_)/(4*256) = 256 blocks.
__global__ void bn_apply_kernel(const float* __restrict__ y, const float* __restrict__ scale,
                                const float* __restrict__ shift, float* __restrict__ xn) {
    int idx = blockIdx.x * 256 + threadIdx.x;     // float4 index
    int gq  = idx & (FQ_ - 1);                    // feature-quad
    float4 v  = reinterpret_cast<const float4*>(y)[idx];
    float4 sc = reinterpret_cast<const float4*>(scale)[gq];
    float4 sh = reinterpret_cast<const float4*>(shift)[gq];
    float4 r;
    r.x = lrelu(v.x * sc.x + sh.x);
    r.y = lrelu(v.y * sc.y + sh.y);
    r.z = lrelu(v.z * sc.z + sh.z);
    r.w = lrelu(v.w * sc.w + sh.w);
    reinterpret_cast<float4*>(xn)[idx] = r;
}

// ---------------------------------------------------------------------------
extern "C" void kernel_launch(void* const* d_in, const int* in_sizes, int n_in,
                              void* d_out, int out_size, void* d_ws, size_t ws_size,
                              hipStream_t stream) {
    (void)in_sizes; (void)n_in; (void)out_size; (void)ws_size;
    const float* x0 = (const float*)d_in[0];   // [T,N,F]
    const float* E  = (const float*)d_in[1];   // [T,N,N,F]
    const float* ew = (const float*)d_in[2];   // [L,3]
    const float* eb = (const float*)d_in[3];   // [L]
    const float* cw = (const float*)d_in[4];   // [L,2]
    const float* nw = (const float*)d_in[5];   // [L,F,F]
    const float* gm = (const float*)d_in[6];   // [L,F]
    const float* bt = (const float*)d_in[7];   // [L,F]
    float* out = (float*)d_out;

    float* ws = (float*)d_ws;
    const size_t TNF = (size_t)ROWS * F_;      // 262144 floats
    float* x1    = ws;
    float* x2    = ws + TNF;
    float* aggr  = ws + 2 * TNF;
    float* ybuf  = ws + 3 * TNF;
    float* nrm   = ws + 4 * TNF;               // ROWS floats
    float* scale = nrm + ROWS;                 // F_ floats
    float* shift = scale + F_;                 // F_ floats

    const size_t OUT1 = (size_t)T_ * 2 * N_ * N_;   // 524288 per output tensor

    // outs[0]: sim of input node feats
    row_norm_kernel<<<ROWS / 8, 256, 0, stream>>>(x0, nrm);
    sim_wmma_kernel<<<128, 256, 0, stream>>>(x0, nrm, out);

    // ---- layer 0 ----
    edge_aggr_kernel<<<ROWS, 256, 0, stream>>>(x0, x0, E, ew, eb, aggr, 0);
    node_gemm_kernel<<<128, 256, 0, stream>>>(x0, aggr, cw, nw, ybuf, 0);
    bn_stats_kernel<<<F_, 256, 0, stream>>>(ybuf, gm, bt, scale, shift, 0);
    bn_apply_kernel<<<(int)(TNF / 1024), 256, 0, stream>>>(ybuf, scale, shift, x1);
    row_norm_kernel<<<ROWS / 8, 256, 0, stream>>>(x1, nrm);
    sim_wmma_kernel<<<128, 256, 0, stream>>>(x1, nrm, out + OUT1);

    // ---- layer 1 (recomputes e0 from E on the fly) ----
    edge_aggr_kernel<<<ROWS, 256, 0, stream>>>(x0, x1, E, ew, eb, aggr, 1);
    node_gemm_kernel<<<128, 256, 0, stream>>>(x1, aggr, cw, nw, ybuf, 1);
    bn_stats_kernel<<<F_, 256, 0, stream>>>(ybuf, gm, bt, scale, shift, 1);
    bn_apply_kernel<<<(int)(TNF / 1024), 256, 0, stream>>>(ybuf, scale, shift, x2);
    row_norm_kernel<<<ROWS / 8, 256, 0, stream>>>(x2, nrm);
    sim_wmma_kernel<<<128, 256, 0, stream>>>(x2, nrm, out + 2 * OUT1);
}